// GIN_12936441496234
// MI455X (gfx1250) — compile-verified
//
#include <hip/hip_runtime.h>
#include <hip/hip_bf16.h>

// GIN forward on MI455X (gfx1250, wave32).
// Memory-bound on edge scatter/gather (~2.5 GB @ 23.3 TB/s); GEMMs (3.8 GFLOP)
// done in exact fp32 via v_wmma_f32_16x16x4_f32 to match the fp32 reference.

#define N_NODES 50000
#define N_EDGES 800000
#define D_FEAT  128
#define N_CLS   40

typedef __attribute__((ext_vector_type(2))) float v2f;
typedef __attribute__((ext_vector_type(8))) float v8f;

// ---------------- degree: deg[v] = #incoming edges ----------------
__global__ void __launch_bounds__(256) gin_deg_kernel(
    const int* __restrict__ dst, float* __restrict__ deg, int nEdges)
{
    int e = blockIdx.x * 256 + threadIdx.x;
    if (e < nEdges) unsafeAtomicAdd(&deg[dst[e]], 1.0f);
}

// ---------------- pad W3[128x40] -> Wp[128x48], b3[40] -> bp[48] ----------------
__global__ void __launch_bounds__(256) gin_pad3_kernel(
    const float* __restrict__ W3, const float* __restrict__ b3,
    float* __restrict__ Wp, float* __restrict__ bp)
{
    int i = blockIdx.x * 256 + threadIdx.x;
    if (i < D_FEAT * 48) {
        int k = i / 48, n = i % 48;
        Wp[i] = (n < N_CLS) ? W3[k * N_CLS + n] : 0.0f;
    }
    if (i < 48) bp[i] = (i < N_CLS) ? b3[i] : 0.0f;
}

// ---------------- scatter: agg[dst] += feat[src] ----------------
// One wave32 per edge; e is wave-uniform -> readfirstlane makes the index
// loads scalar (s_load). Each lane gathers a float4 (coalesced 512B/edge)
// and issues 4 native global_atomic_add_f32.
__global__ void __launch_bounds__(256) gin_scatter_kernel(
    const float* __restrict__ feat, const int* __restrict__ src,
    const int* __restrict__ dst, float* __restrict__ agg, int nEdges)
{
    int tid = blockIdx.x * 256 + threadIdx.x;
    int e = __builtin_amdgcn_readfirstlane(tid >> 5);   // wave-uniform edge id
    if (e >= nEdges) return;
    int s = src[e];                                     // scalar loads
    int d = dst[e];
    int q = (threadIdx.x & 31) << 2;                    // feature offset 0..124
    const float4 v = *(const float4*)(feat + (size_t)s * D_FEAT + q);
    float* o = agg + (size_t)d * D_FEAT + q;
    unsafeAtomicAdd(o + 0, v.x);
    unsafeAtomicAdd(o + 1, v.y);
    unsafeAtomicAdd(o + 2, v.z);
    unsafeAtomicAdd(o + 3, v.w);
}

// ---------------- fused GIN layer GEMM ----------------
// out[m, 0..N) = act( (X[m,:] + agg[m,:] * scale_m) @ W + b )
// N is compile-time; W/bias are padded to NP = ceil16(N) so the inner loop
// has no conditionals. Block = 256 thr = 8 wave32; block owns a 16-row M-tile
// (Z staged once in LDS), wave w owns cols [16w,16w+16); K=128 in 32 steps of
// v_wmma_f32_16x16x4_f32 with immediate-offset ds_load_b64 / global_load_b32.
template <int N, bool RELU, bool MEAN>
__global__ void __launch_bounds__(256) gin_gemm_kernel(
    const float* __restrict__ X, const float* __restrict__ agg,
    const float* __restrict__ deg,          // used iff MEAN
    const float* __restrict__ W,            // [128 x NP] row-major, zero-padded
    const float* __restrict__ bias,         // [NP], zero-padded
    float* __restrict__ out)                // [M x N]
{
    constexpr int NP = (N + 15) & ~15;
    constexpr int NT = NP / 16;
    __shared__ float zt[16][132];           // pad stride 132 -> conflict-free b64 reads
    const int m0 = blockIdx.x * 16;
    const int t  = threadIdx.x;

    // Stage Z = X + agg*scale  (16 x 128, 8 elems/thread)
    for (int i = t; i < 16 * D_FEAT; i += 256) {
        int r = i >> 7, c = i & 127;
        size_t g = (size_t)(m0 + r) * D_FEAT + c;
        float s = 1.0f;
        if (MEAN) s = 1.0f / fmaxf(deg[m0 + r], 1.0f);
        zt[r][c] = X[g] + agg[g] * s;
    }
    __syncthreads();

    const int wave = t >> 5;
    if (wave >= NT) return;                 // whole-wave exit -> WMMA EXEC all-1s
    const int lane = t & 31;
    const int half = lane >> 4;             // ISA A/B layout: VGPR0=K0/K2, VGPR1=K1/K3
    const int lr   = lane & 15;
    const int col  = wave * 16 + lr;

    const float* wp = W + (size_t)(2 * half) * NP + col;  // B base (global, L2-hot)
    const float* zp = &zt[lr][2 * half];                  // A base (LDS)

    v8f acc = {0.f, 0.f, 0.f, 0.f, 0.f, 0.f, 0.f, 0.f};
#pragma unroll
    for (int k0 = 0; k0 < D_FEAT; k0 += 4) {
        v2f A; A.x = zp[k0];            A.y = zp[k0 + 1];
        v2f B; B.x = wp[(size_t)k0 * NP]; B.y = wp[(size_t)(k0 + 1) * NP];
        acc = __builtin_amdgcn_wmma_f32_16x16x4_f32(
            /*neg_a=*/false, A, /*neg_b=*/false, B,
            /*c_mod=*/(short)0, acc, /*reuse_a=*/false, /*reuse_b=*/false);
    }

    // C/D layout: acc[j] on lane -> row m0 + j + 8*half, col
    const float bv = bias[col];             // padded -> always in-bounds
#pragma unroll
    for (int j = 0; j < 8; ++j) {
        int row = m0 + j + 8 * half;
        float v = acc[j] + bv;
        if (RELU) v = fmaxf(v, 0.0f);
        if (N % 16 == 0 || col < N)         // folds away when N is 16-aligned
            out[(size_t)row * N + col] = v;
    }
}

extern "C" void kernel_launch(void* const* d_in, const int* in_sizes, int n_in,
                              void* d_out, int out_size, void* d_ws, size_t ws_size,
                              hipStream_t stream) {
    const float* x   = (const float*)d_in[0];
    const int*   src = (const int*)  d_in[1];
    const int*   dst = (const int*)  d_in[2];
    const float* W1  = (const float*)d_in[3];
    const float* b1  = (const float*)d_in[4];
    const float* W2  = (const float*)d_in[5];
    const float* b2  = (const float*)d_in[6];
    const float* W3  = (const float*)d_in[7];
    const float* b3  = (const float*)d_in[8];
    float* out = (float*)d_out;

    const size_t featElems = (size_t)N_NODES * D_FEAT;
    float* agg = (float*)d_ws;              // 25.6 MB
    float* h   = agg + featElems;           // 25.6 MB
    float* deg = h + featElems;             // 200 KB
    float* W3p = deg + N_NODES;             // 24 KB  (128 x 48)
    float* b3p = W3p + D_FEAT * 48;         // 192 B

    const int scatterBlocks = (N_EDGES * 32 + 255) / 256;   // 100000
    const int degBlocks     = (N_EDGES + 255) / 256;
    const int gemmBlocks    = N_NODES / 16;                 // 3125 exactly

    // degree + padded layer-3 weights (computed once)
    hipMemsetAsync(deg, 0, (size_t)N_NODES * sizeof(float), stream);
    gin_deg_kernel<<<degBlocks, 256, 0, stream>>>(dst, deg, N_EDGES);
    gin_pad3_kernel<<<(D_FEAT * 48 + 255) / 256, 256, 0, stream>>>(W3, b3, W3p, b3p);

    // layer 1: h = relu((x + sum_agg(x)) @ W1 + b1)
    hipMemsetAsync(agg, 0, featElems * sizeof(float), stream);
    gin_scatter_kernel<<<scatterBlocks, 256, 0, stream>>>(x, src, dst, agg, N_EDGES);
    gin_gemm_kernel<D_FEAT, true, false><<<gemmBlocks, 256, 0, stream>>>(
        x, agg, nullptr, W1, b1, h);

    // layer 2: h = relu((h + sum_agg(h)) @ W2 + b2)  (in-place safe: block-local rows)
    hipMemsetAsync(agg, 0, featElems * sizeof(float), stream);
    gin_scatter_kernel<<<scatterBlocks, 256, 0, stream>>>(h, src, dst, agg, N_EDGES);
    gin_gemm_kernel<D_FEAT, true, false><<<gemmBlocks, 256, 0, stream>>>(
        h, agg, nullptr, W2, b2, h);

    // layer 3: out = (h + mean_agg(h)) @ W3 + b3
    hipMemsetAsync(agg, 0, featElems * sizeof(float), stream);
    gin_scatter_kernel<<<scatterBlocks, 256, 0, stream>>>(h, src, dst, agg, N_EDGES);
    gin_gemm_kernel<N_CLS, false, true><<<gemmBlocks, 256, 0, stream>>>(
        h, agg, deg, W3p, b3p, out);
}